// GCN_47304769798728
// MI455X (gfx1250) — compile-verified
//
#include <hip/hip_runtime.h>
#include <hip/hip_bf16.h>

typedef __attribute__((ext_vector_type(2))) float v2f;
typedef __attribute__((ext_vector_type(8))) float v8f;

#define D_IN 128
#define BN_EPS 1e-5f

// ---------------------------------------------------------------------------
// Degree / normalization
// ---------------------------------------------------------------------------
__global__ void deg_init_kernel(int* __restrict__ degi, int n) {
    int i = blockIdx.x * blockDim.x + threadIdx.x;
    if (i < n) degi[i] = 1;  // self-loop contributes 1
}

__global__ void deg_count_kernel(const int* __restrict__ dst, int* __restrict__ degi, int E) {
    int e = blockIdx.x * blockDim.x + threadIdx.x;
    if (e < E) atomicAdd(&degi[dst[e]], 1);
}

__global__ void dinv_kernel(float* __restrict__ dinv, int n) {
    int i = blockIdx.x * blockDim.x + threadIdx.x;
    if (i < n) {
        int c = ((const int*)dinv)[i];      // in-place: read int degree
        dinv[i] = rsqrtf((float)c);         // write float dinv (deg >= 1 always)
    }
}

// ---------------------------------------------------------------------------
// fp32 WMMA GEMM: H[N x DOUT] = X[N x 128] @ W[128 x DOUT]
// One wave per 16x16 output tile; K=128 via 32x v_wmma_f32_16x16x4_f32.
// blockDim.x = (DOUT/16)*32, gridDim.x = N/16  (N % 16 == 0)
// ---------------------------------------------------------------------------
template <int DOUT>
__global__ void gemm_wmma_kernel(const float* __restrict__ X,
                                 const float* __restrict__ W,
                                 float* __restrict__ H) {
    const int lane = threadIdx.x & 31;
    const int wave = threadIdx.x >> 5;
    const int row0 = blockIdx.x << 4;
    const int col0 = wave << 4;
    const int half = lane >> 4;     // 0: K pair {k,k+1}; 1: K pair {k+2,k+3}
    const int l    = lane & 15;

    v8f acc = {};
    const float* xrow = X + (size_t)(row0 + l) * D_IN;  // A: M = l (both halves)

#pragma unroll
    for (int k = 0; k < D_IN; k += 4) {
        const int ka = k + half * 2;
        v2f a, b;
        // A fragment 16x4: VGPR0=K(ka), VGPR1=K(ka+1), M striped over lanes 0-15
        a.x = xrow[ka];
        a.y = xrow[ka + 1];
        // B fragment 4x16: row K across lanes (N = col0 + l), halves split K
        b.x = W[(size_t)ka * DOUT + col0 + l];
        b.y = W[(size_t)(ka + 1) * DOUT + col0 + l];
        acc = __builtin_amdgcn_wmma_f32_16x16x4_f32(
            /*neg_a=*/false, a, /*neg_b=*/false, b,
            /*c_mod=*/(short)0, acc, /*reuse_a=*/false, /*reuse_b=*/false);
    }

    // C/D 16x16: VGPR r holds row (half*8 + r), N striped over lanes 0-15
#pragma unroll
    for (int r = 0; r < 8; ++r)
        H[(size_t)(row0 + half * 8 + r) * DOUT + col0 + l] = acc[r];
}

// ---------------------------------------------------------------------------
// Broadcast bias into aggregation buffer (float4): AGG[i, f] = b[f]
// total4 = N*D/4 float4 elements
// ---------------------------------------------------------------------------
template <int D>
__global__ void bias_init_kernel(float4* __restrict__ AGG, const float4* __restrict__ b,
                                 unsigned total4) {
    unsigned i = blockIdx.x * blockDim.x + threadIdx.x;
    if (i < total4) AGG[i] = b[i & (D / 4 - 1)];
}

// ---------------------------------------------------------------------------
// Edge-parallel gather/scale/scatter (float4 gather, scalar fp32 atomics):
//   AGG[dst, :] += H[src, :] * dinv[src] * dinv[dst]
// edges [0,E) from edge_index; edges [E, E+N) are self-loops.
// One thread per (edge, feature-quad); D/4 lanes of a wave share one edge,
// so H reads are global_load_b128 coalesced along the feature dimension.
// ---------------------------------------------------------------------------
template <int D>
__global__ void scatter_kernel(const float* __restrict__ H,
                               const float* __restrict__ dinv,
                               const int* __restrict__ src,
                               const int* __restrict__ dst,
                               float* __restrict__ AGG, int E, int N) {
    constexpr int Q = D / 4;  // quads per edge (32 or 16), power of two
    unsigned tid = blockIdx.x * blockDim.x + threadIdx.x;
    unsigned e = tid / Q;
    unsigned q = tid & (Q - 1);
    if (e >= (unsigned)(E + N)) return;
    int s, d;
    if (e < (unsigned)E) {
        s = src[e];
        d = dst[e];
    } else {
        s = d = (int)(e - (unsigned)E);
    }
    float w = dinv[s] * dinv[d];
    const float4 h = ((const float4*)(H + (size_t)s * D))[q];  // b128 load
    float* outp = AGG + (size_t)d * D + q * 4u;
    unsafeAtomicAdd(outp + 0, h.x * w);  // global_atomic_add_f32 x4
    unsafeAtomicAdd(outp + 1, h.y * w);
    unsafeAtomicAdd(outp + 2, h.z * w);
    unsafeAtomicAdd(outp + 3, h.w * w);
}

// ---------------------------------------------------------------------------
// Fused eval-mode BatchNorm + ReLU, in place, float4.
// ---------------------------------------------------------------------------
template <int D>
__global__ void bn_relu_kernel(float4* __restrict__ A,
                               const float* __restrict__ scale,
                               const float* __restrict__ bias,
                               const float* __restrict__ mean,
                               const float* __restrict__ var,
                               unsigned total4) {
    unsigned i = blockIdx.x * blockDim.x + threadIdx.x;
    if (i >= total4) return;
    unsigned f = (i * 4u) & (D - 1);  // base feature of this quad
    float4 v = A[i];
    float g0 = scale[f + 0] * rsqrtf(var[f + 0] + BN_EPS);
    float g1 = scale[f + 1] * rsqrtf(var[f + 1] + BN_EPS);
    float g2 = scale[f + 2] * rsqrtf(var[f + 2] + BN_EPS);
    float g3 = scale[f + 3] * rsqrtf(var[f + 3] + BN_EPS);
    v.x = fmaxf((v.x - mean[f + 0]) * g0 + bias[f + 0], 0.0f);
    v.y = fmaxf((v.y - mean[f + 1]) * g1 + bias[f + 1], 0.0f);
    v.z = fmaxf((v.z - mean[f + 2]) * g2 + bias[f + 2], 0.0f);
    v.w = fmaxf((v.w - mean[f + 3]) * g3 + bias[f + 3], 0.0f);
    A[i] = v;
}

// ---------------------------------------------------------------------------
// Host-side launcher
// ---------------------------------------------------------------------------
extern "C" void kernel_launch(void* const* d_in, const int* in_sizes, int n_in,
                              void* d_out, int out_size, void* d_ws, size_t ws_size,
                              hipStream_t stream) {
    const float* x       = (const float*)d_in[0];
    const int*   eidx    = (const int*)d_in[1];
    const float* W1      = (const float*)d_in[2];
    const float* b1      = (const float*)d_in[3];
    const float* bn1_s   = (const float*)d_in[4];
    const float* bn1_b   = (const float*)d_in[5];
    const float* bn1_m   = (const float*)d_in[6];
    const float* bn1_v   = (const float*)d_in[7];
    const float* W2      = (const float*)d_in[8];
    const float* b2      = (const float*)d_in[9];
    const float* bn2_s   = (const float*)d_in[10];
    const float* bn2_b   = (const float*)d_in[11];
    const float* bn2_m   = (const float*)d_in[12];
    const float* bn2_v   = (const float*)d_in[13];
    const float* W3      = (const float*)d_in[14];
    const float* b3      = (const float*)d_in[15];
    float* out = (float*)d_out;

    const int N = in_sizes[0] / D_IN;     // 100000
    const int E = in_sizes[1] / 2;        // 1600000
    const int* src = eidx;                // edge_index[0]
    const int* dst = eidx + E;            // edge_index[1]

    // Workspace layout: dinv | H (N x 128) | AGG (N x 128)
    float* dinv = (float*)d_ws;
    float* H    = dinv + N;
    float* AGG  = H + (size_t)N * D_IN;

    const int T = 256;
    const unsigned q128 = (unsigned)N * 32u;   // N*128/4 quads
    const unsigned q64  = (unsigned)N * 16u;   // N*64/4  quads
    const unsigned sc128 = (unsigned)(((long long)(E + N) * 32 + T - 1) / T);
    const unsigned sc64  = (unsigned)(((long long)(E + N) * 16 + T - 1) / T);
    const int mTiles = N / 16;            // N divisible by 16

    // --- normalization: deg -> dinv ---
    deg_init_kernel<<<(N + T - 1) / T, T, 0, stream>>>((int*)dinv, N);
    deg_count_kernel<<<(E + T - 1) / T, T, 0, stream>>>(dst, (int*)dinv, E);
    dinv_kernel<<<(N + T - 1) / T, T, 0, stream>>>(dinv, N);

    // --- layer 1: GEMM -> bias-init -> scatter -> BN/ReLU ---
    gemm_wmma_kernel<128><<<mTiles, 8 * 32, 0, stream>>>(x, W1, H);
    bias_init_kernel<128><<<(q128 + T - 1) / T, T, 0, stream>>>((float4*)AGG, (const float4*)b1, q128);
    scatter_kernel<128><<<sc128, T, 0, stream>>>(H, dinv, src, dst, AGG, E, N);
    bn_relu_kernel<128><<<(q128 + T - 1) / T, T, 0, stream>>>((float4*)AGG, bn1_s, bn1_b, bn1_m, bn1_v, q128);

    // --- layer 2 ---
    gemm_wmma_kernel<128><<<mTiles, 8 * 32, 0, stream>>>(AGG, W2, H);
    bias_init_kernel<128><<<(q128 + T - 1) / T, T, 0, stream>>>((float4*)AGG, (const float4*)b2, q128);
    scatter_kernel<128><<<sc128, T, 0, stream>>>(H, dinv, src, dst, AGG, E, N);
    bn_relu_kernel<128><<<(q128 + T - 1) / T, T, 0, stream>>>((float4*)AGG, bn2_s, bn2_b, bn2_m, bn2_v, q128);

    // --- layer 3: GEMM (N x 64) -> scatter into d_out ---
    gemm_wmma_kernel<64><<<mTiles, 4 * 32, 0, stream>>>(AGG, W3, H);
    bias_init_kernel<64><<<(q64 + T - 1) / T, T, 0, stream>>>((float4*)out, (const float4*)b3, q64);
    scatter_kernel<64><<<sc64, T, 0, stream>>>(H, dinv, src, dst, out, E, N);
}